// WhiteningAndColoring_81561428951818
// MI455X (gfx1250) — compile-verified
//
#include <hip/hip_runtime.h>
#include <math.h>

// ---------------------------------------------------------------------------
// WCT (whitening & coloring) for MI455X / gfx1250, wave32, f32 WMMA 16x16x4.
// SVD replaced by coupled Newton-Schulz (pure GEMM -> v_wmma_f32_16x16x4_f32).
// Covariance SYRK uses LDS tiling fed by global_load_async_to_lds_b128.
// ---------------------------------------------------------------------------

typedef float v2f __attribute__((ext_vector_type(2)));
typedef float v8f __attribute__((ext_vector_type(8)));

#define WCT_EPS   1e-8f
#define WCT_ALPHA 0.8f
#define NS_ITERS  20
#define KC        16          // K rows staged per chunk
#define LDS_PITCH 68          // 64 + 4 pad floats: half-waves hit disjoint banks

// ---------------- small utility kernels ----------------

__global__ void zero_kernel(float* __restrict__ p, int n) {
    int i = blockIdx.x * blockDim.x + threadIdx.x;
    if (i < n) p[i] = 0.0f;
}

// accumulate per-channel sums; grid = (C/256, NCHUNKS), block = 256
__global__ void mean_acc_kernel(const float* __restrict__ x, float* __restrict__ mean,
                                int N, int C) {
    int c  = blockIdx.x * blockDim.x + threadIdx.x;
    int per = N / gridDim.y;
    int n0 = blockIdx.y * per;
    float s = 0.0f;
    for (int n = n0; n < n0 + per; ++n) s += x[(size_t)n * C + c];
    atomicAdd(&mean[c], s);
}

__global__ void scale_kernel(float* __restrict__ p, float f, int n) {
    int i = blockIdx.x * blockDim.x + threadIdx.x;
    if (i < n) p[i] *= f;
}

__global__ void add_eps_diag_kernel(float* __restrict__ cov, int C, float eps) {
    int i = blockIdx.x * blockDim.x + threadIdx.x;
    if (i < C) cov[(size_t)i * C + i] += eps;
}

// sum of squares (Frobenius^2) with wave reduction -> atomicAdd per wave
__global__ void sumsq_kernel(const float* __restrict__ A, float* __restrict__ out, int n) {
    int i = blockIdx.x * blockDim.x + threadIdx.x;
    int stride = gridDim.x * blockDim.x;
    float v = 0.0f;
    for (int j = i; j < n; j += stride) { float a = A[j]; v += a * a; }
    for (int off = 16; off > 0; off >>= 1) v += __shfl_down(v, off, 32);
    if ((threadIdx.x & 31) == 0) atomicAdd(out, v);
}

// Y = A / ||A||_F , Z = I
__global__ void ns_init_kernel(const float* __restrict__ A, const float* __restrict__ ssq,
                               float* __restrict__ Y, float* __restrict__ Z, int C) {
    int i = blockIdx.x * blockDim.x + threadIdx.x;
    int n = C * C;
    if (i >= n) return;
    float inv_norm = rsqrtf(*ssq);        // 1/||A||_F
    Y[i] = A[i] * inv_norm;
    int r = i / C, c = i - r * C;
    Z[i] = (r == c) ? 1.0f : 0.0f;
}

// combined scalar for M = color @ whiten = (ssq_s/ssq_c)^(1/4) * Ys @ Zc
__global__ void sfac_kernel(const float* __restrict__ ssq_s,
                            const float* __restrict__ ssq_c,
                            float* __restrict__ sfac) {
    if (threadIdx.x == 0 && blockIdx.x == 0)
        *sfac = sqrtf(sqrtf(*ssq_s / *ssq_c));
}

// ---------------- WMMA GEMM kernels ----------------
// Fragment layouts per CDNA5 ISA 7.12.2 (wave32):
//   A (16x4 f32): lane l -> row m = l&15; VGPR pair holds K = 2*(l>>4) + {0,1}
//   B (4x16 f32): lane l -> col n = l&15; VGPR pair holds K = 2*(l>>4) + {0,1}
//   C/D (16x16)  : lane l, vgpr v -> (row = v + 8*(l>>4), col = l&15)

// Partial SYRK: cov += (X-m)(X-m)^T * inv_nm1 over this block's K-range.
// Block (256 thr, 8 waves) owns a 64x64 output region; each wave computes two
// 16x16 tiles. Operand strips are staged to LDS via async global->LDS b128.
__global__ void cov_syrk_lds_kernel(const float* __restrict__ x,
                                    const float* __restrict__ mean,
                                    float* __restrict__ cov, int N, int C,
                                    float inv_nm1) {
    __shared__ float ldsA[KC * LDS_PITCH];
    __shared__ float ldsB[KC * LDS_PITCH];

    const int tid  = threadIdx.x;
    const int lane = tid & 31;
    const int wave = tid >> 5;                 // 0..7
    const int bi   = blockIdx.x >> 3;          // C/64 = 8 block rows
    const int bj   = blockIdx.x & 7;
    const int ci0  = bi << 6, cj0 = bj << 6;

    const int kchunks = N / (KC * gridDim.y);
    const int k0      = blockIdx.y * kchunks * KC;

    // wave -> subtile (ti, tj0) and (ti, tj0+1) in the 4x4 subtile grid
    const int ti  = wave >> 1;
    const int tj0 = (wave & 1) << 1;

    const int m    = lane & 15;
    const int half = lane >> 4;
    const int kk   = half << 1;

    const float am  = mean[ci0 + (ti  << 4) + m];
    const float bm0 = mean[cj0 + (tj0 << 4) + m];
    const float bm1 = mean[cj0 + ((tj0 + 1) << 4) + m];

    v8f acc0 = {}, acc1 = {};

    // staging: thread t copies 16B: row = t/16 (0..15), colgroup = (t%16)*4
    const int srow = tid >> 4;
    const int scol = (tid & 15) << 2;
    const unsigned ldsA_off = (unsigned)(uintptr_t)&ldsA[srow * LDS_PITCH + scol];
    const unsigned ldsB_off = (unsigned)(uintptr_t)&ldsB[srow * LDS_PITCH + scol];

    for (int c = 0; c < kchunks; ++c) {
        const int kbase = k0 + c * KC;
        unsigned long long gA =
            (unsigned long long)(x + (size_t)(kbase + srow) * C + ci0 + scol);
        unsigned long long gB =
            (unsigned long long)(x + (size_t)(kbase + srow) * C + cj0 + scol);
        asm volatile("global_load_async_to_lds_b128 %0, %1, off"
                     :: "v"(ldsA_off), "v"(gA) : "memory");
        asm volatile("global_load_async_to_lds_b128 %0, %1, off"
                     :: "v"(ldsB_off), "v"(gB) : "memory");
        asm volatile("s_wait_asynccnt 0x0" ::: "memory");
        __syncthreads();

        for (int ks = 0; ks < KC; ks += 4) {
            v2f a, b0, b1;
            a.x  = ldsA[(ks + kk)     * LDS_PITCH + (ti  << 4) + m] - am;
            a.y  = ldsA[(ks + kk + 1) * LDS_PITCH + (ti  << 4) + m] - am;
            b0.x = ldsB[(ks + kk)     * LDS_PITCH + (tj0 << 4) + m] - bm0;
            b0.y = ldsB[(ks + kk + 1) * LDS_PITCH + (tj0 << 4) + m] - bm0;
            b1.x = ldsB[(ks + kk)     * LDS_PITCH + ((tj0 + 1) << 4) + m] - bm1;
            b1.y = ldsB[(ks + kk + 1) * LDS_PITCH + ((tj0 + 1) << 4) + m] - bm1;
            acc0 = __builtin_amdgcn_wmma_f32_16x16x4_f32(false, a, false, b0,
                                                         (short)0, acc0, false, false);
            acc1 = __builtin_amdgcn_wmma_f32_16x16x4_f32(false, a, false, b1,
                                                         (short)0, acc1, false, false);
        }
        __syncthreads();
    }

    for (int v = 0; v < 8; ++v) {
        int row  = ci0 + (ti << 4) + v + 8 * half;
        int col0 = cj0 + (tj0 << 4) + m;
        atomicAdd(&cov[(size_t)row * C + col0],      acc0[v] * inv_nm1);
        atomicAdd(&cov[(size_t)row * C + col0 + 16], acc1[v] * inv_nm1);
    }
}

// D = alpha * s * (A @ B) + beta * I   (512x512x512, row-major)
__global__ void gemm512_kernel(const float* __restrict__ A, const float* __restrict__ B,
                               float* __restrict__ D, float alpha, float beta,
                               const float* __restrict__ sptr) {
    const int C = 512;
    int wave = threadIdx.x >> 5;
    int lane = threadIdx.x & 31;
    int tile = blockIdx.x * (blockDim.x >> 5) + wave;
    int ti = tile >> 5, tj = tile & 31;
    int ci = ti << 4, cj = tj << 4;
    int m = lane & 15, half = lane >> 4, kk = half << 1;
    float s = alpha * (sptr ? *sptr : 1.0f);
    v8f acc = {};
    const float* arow = A + (size_t)(ci + m) * C + kk;   // contiguous K-pair
    for (int k = 0; k < C; k += 4) {
        v2f a = *(const v2f*)(arow + k);
        v2f b;
        b.x = B[(size_t)(k + kk)     * C + cj + m];
        b.y = B[(size_t)(k + kk + 1) * C + cj + m];
        acc = __builtin_amdgcn_wmma_f32_16x16x4_f32(false, a, false, b,
                                                    (short)0, acc, false, false);
    }
    for (int v = 0; v < 8; ++v) {
        int row = ci + v + 8 * half;
        int col = cj + m;
        float val = acc[v] * s + ((row == col) ? beta : 0.0f);
        D[(size_t)row * C + col] = val;
    }
}

// out[n,c] = ALPHA*((M @ fc)[c,n] + ms[c]) + (1-ALPHA)*content[n,c]
// fc[k,n] = content[n,k] - mc[k].   One wave per (16 chan x 16 pixel) tile.
__global__ void apply_kernel(const float* __restrict__ M, const float* __restrict__ content,
                             const float* __restrict__ mc, const float* __restrict__ ms,
                             float* __restrict__ out, int N, int C) {
    int wave = threadIdx.x >> 5;
    int lane = threadIdx.x & 31;
    int tile = blockIdx.x * (blockDim.x >> 5) + wave;
    int ntj  = N >> 4;                       // pixel tiles
    int ti = tile / ntj, tj = tile % ntj;
    int c0 = ti << 4, n0 = tj << 4;
    int m = lane & 15, half = lane >> 4, kk = half << 1;
    v8f acc = {};
    const float* arow = M + (size_t)(c0 + m) * C + kk;
    const float* crow = content + (size_t)(n0 + m) * C + kk; // pixel row for B
    for (int k = 0; k < C; k += 4) {
        v2f a = *(const v2f*)(arow + k);
        v2f b = *(const v2f*)(crow + k);
        b.x -= mc[k + kk];
        b.y -= mc[k + kk + 1];
        acc = __builtin_amdgcn_wmma_f32_16x16x4_f32(false, a, false, b,
                                                    (short)0, acc, false, false);
    }
    int n = n0 + m;
    for (int v = 0; v < 8; ++v) {
        int c = c0 + v + 8 * half;
        size_t idx = (size_t)n * C + c;
        out[idx] = WCT_ALPHA * (acc[v] + ms[c]) + (1.0f - WCT_ALPHA) * content[idx];
    }
}

// ---------------- host orchestration ----------------

extern "C" void kernel_launch(void* const* d_in, const int* in_sizes, int n_in,
                              void* d_out, int out_size, void* d_ws, size_t ws_size,
                              hipStream_t stream) {
    const float* content = (const float*)d_in[0];
    const float* style   = (const float*)d_in[1];
    float* out = (float*)d_out;

    const int C  = 512;
    const int N  = in_sizes[0] / C;          // 16384
    const int CC = C * C;                    // 262144

    float* w      = (float*)d_ws;
    float* mean_c = w;                       // 512
    float* mean_s = w + 512;                 // 512
    float* ssq_c  = w + 1024;
    float* ssq_s  = w + 1025;
    float* sfac   = w + 1026;
    float* cov_c  = w + 2048;                // 512^2 (reused as M at the end)
    float* cov_s  = cov_c + CC;
    float* B0 = cov_s + CC;                  // NS ping-pong + T + saved Zc
    float* B1 = B0 + CC;
    float* B2 = B1 + CC;                     // T
    float* B3 = B2 + CC;
    float* B4 = B3 + CC;
    float* Zc = B4 + CC;                     // saved content whitening matrix

    // 1) zero means/scalars and both covariance accumulators
    zero_kernel<<<(1027 + 255) / 256, 256, 0, stream>>>(w, 1027);
    zero_kernel<<<(2 * CC + 255) / 256, 256, 0, stream>>>(cov_c, 2 * CC);

    // 2) per-channel means
    {
        dim3 g(C / 256, 64);
        mean_acc_kernel<<<g, 256, 0, stream>>>(content, mean_c, N, C);
        mean_acc_kernel<<<g, 256, 0, stream>>>(style,   mean_s, N, C);
        scale_kernel<<<(1024 + 255) / 256, 256, 0, stream>>>(mean_c, 1.0f / (float)N, 1024);
    }

    // 3) covariances: LDS-tiled SYRK, K split over grid.y, atomic partials
    {
        float inv_nm1 = 1.0f / ((float)N - 1.0f);
        dim3 g(64, 16);  // 8x8 block tiles of 64x64, 16-way K split
        cov_syrk_lds_kernel<<<g, 256, 0, stream>>>(content, mean_c, cov_c, N, C, inv_nm1);
        cov_syrk_lds_kernel<<<g, 256, 0, stream>>>(style,   mean_s, cov_s, N, C, inv_nm1);
        add_eps_diag_kernel<<<2, 256, 0, stream>>>(cov_c, C, WCT_EPS);
        add_eps_diag_kernel<<<2, 256, 0, stream>>>(cov_s, C, WCT_EPS);
    }

    // 4) Frobenius norms^2 + combined scale factor
    sumsq_kernel<<<256, 256, 0, stream>>>(cov_c, ssq_c, CC);
    sumsq_kernel<<<256, 256, 0, stream>>>(cov_s, ssq_s, CC);
    sfac_kernel<<<1, 32, 0, stream>>>(ssq_s, ssq_c, sfac);

    // 5) Newton-Schulz on content covariance: Z -> Cov_c^{-1/2} * ||Cov_c||_F^{1/2}
    {
        ns_init_kernel<<<(CC + 255) / 256, 256, 0, stream>>>(cov_c, ssq_c, B0, B1, C);
        float *Y = B0, *Z = B1, *T = B2, *Yn = B3, *Zn = B4;
        for (int it = 0; it < NS_ITERS; ++it) {
            gemm512_kernel<<<128, 256, 0, stream>>>(Z, Y, T, -1.0f, 3.0f, nullptr); // T = 3I - Z@Y
            gemm512_kernel<<<128, 256, 0, stream>>>(Y, T, Yn, 0.5f, 0.0f, nullptr); // Y' = .5 Y@T
            gemm512_kernel<<<128, 256, 0, stream>>>(T, Z, Zn, 0.5f, 0.0f, nullptr); // Z' = .5 T@Z
            float* t0 = Y; Y = Yn; Yn = t0;
            float* t1 = Z; Z = Zn; Zn = t1;
        }
        hipMemcpyAsync(Zc, Z, (size_t)CC * sizeof(float), hipMemcpyDeviceToDevice, stream);
    }

    // 6) Newton-Schulz on style covariance: Y -> Cov_s^{1/2} / ||Cov_s||_F^{1/2}
    float* Ys;
    {
        ns_init_kernel<<<(CC + 255) / 256, 256, 0, stream>>>(cov_s, ssq_s, B0, B1, C);
        float *Y = B0, *Z = B1, *T = B2, *Yn = B3, *Zn = B4;
        for (int it = 0; it < NS_ITERS; ++it) {
            gemm512_kernel<<<128, 256, 0, stream>>>(Z, Y, T, -1.0f, 3.0f, nullptr);
            gemm512_kernel<<<128, 256, 0, stream>>>(Y, T, Yn, 0.5f, 0.0f, nullptr);
            gemm512_kernel<<<128, 256, 0, stream>>>(T, Z, Zn, 0.5f, 0.0f, nullptr);
            float* t0 = Y; Y = Yn; Yn = t0;
            float* t1 = Z; Z = Zn; Zn = t1;
        }
        Ys = Y;
    }

    // 7) M = color @ whiten = (ssq_s/ssq_c)^(1/4) * Ys @ Zc   (into cov_c buffer)
    float* Mmat = cov_c;
    gemm512_kernel<<<128, 256, 0, stream>>>(Ys, Zc, Mmat, 1.0f, 0.0f, sfac);

    // 8) fused apply + blend:  out = ALPHA*(M@fc + ms) + (1-ALPHA)*content
    {
        int tiles = (C >> 4) * (N >> 4);     // 32 * 1024 = 32768
        apply_kernel<<<tiles / 8, 256, 0, stream>>>(Mmat, content, mean_c, mean_s, out, N, C);
    }
}